// OneLayerMambaEHRModel_90580860272642
// MI455X (gfx1250) — compile-verified
//
#include <hip/hip_runtime.h>

#define BSZ    32
#define DMODEL 64
#define SEQ    2048
#define DINNER 512
#define DSTATE 20
#define DCONV  4
#define DTRANK 4
#define MTOT   (BSZ * SEQ)      // 65536 rows
#define NXZ    1024             // 2 * DINNER
#define NDBC   48               // 44 padded to 48 (3 WMMA n-tiles)

typedef float v2f __attribute__((ext_vector_type(2)));
typedef float v8f __attribute__((ext_vector_type(8)));
typedef unsigned int v4u __attribute__((ext_vector_type(4)));
typedef int v4i __attribute__((ext_vector_type(4)));
typedef int v8i __attribute__((ext_vector_type(8)));

// D = A(16x4 f32) * B(4x16 f32) + C(16x16 f32), wave32, fp32 WMMA path
__device__ __forceinline__ v8f wmma4(v2f a, v2f b, v8f c) {
  return __builtin_amdgcn_wmma_f32_16x16x4_f32(false, a, false, b, (short)0, c,
                                               false, false);
}

// ---------------------------------------------------------------------------
// Tensor Data Mover: DMA a 16-row x 512-col f32 tile (row-major, row stride
// 512 elems) from global memory into LDS, inserting 2 pad dwords after every
// 256 dwords stored.  Padded LDS dword index of element (m,k):
//      idx(m,k) = m*516 + k + ((k>>8)<<1)
// so the per-lane strided reads (fixed k, m = lane) hit banks 4*m mod 64 ->
// conflict-free across a half-wave.  D# layout per CDNA5 ISA.
// ---------------------------------------------------------------------------
__device__ __forceinline__ void tdm_load_tile_16x512(const float* gsrc,
                                                     float* lds_dst) {
  const unsigned long long ga = (unsigned long long)(uintptr_t)gsrc;
  const unsigned int la = (unsigned int)(uintptr_t)lds_dst; // low 32b = LDS off
  v4u g0;
  g0[0] = 1u;                                   // count=1, user descriptor
  g0[1] = la;                                   // lds_addr (bytes)
  g0[2] = (unsigned int)(ga & 0xffffffffu);     // global_addr[31:0]
  g0[3] = (unsigned int)((ga >> 32) & 0x01ffffffu) | (2u << 30); // addr hi|type=2
  v8i g1;
  g1[0] = (int)((2u << 16)        // data_size = 4 bytes
              | (1u << 20)        // pad_enable
              | (7u << 22)        // pad_interval code 7 -> every 256 dwords
              | (1u << 25));      // pad_amount  code 1 -> 2 dwords
  g1[1] = (int)(512u << 16);      // tensor_dim0[15:0] = 512 (elements)
  g1[2] = (int)(16u << 16);       // tensor_dim0 hi = 0 | tensor_dim1 lo = 16
  g1[3] = (int)(512u << 16);      // tensor_dim1 hi = 0 | tile_dim0 = 512
  g1[4] = 16;                     // tile_dim1 = 16 | tile_dim2 = 0
  g1[5] = 512;                    // tensor_dim0_stride lo = 512 elements
  g1[6] = 0;                      // stride hi | tensor_dim1_stride lo
  g1[7] = 0;
  v4i z4 = {0, 0, 0, 0};
#if __clang_major__ >= 23
  v8i z8 = {0, 0, 0, 0, 0, 0, 0, 0};
  __builtin_amdgcn_tensor_load_to_lds(g0, g1, z4, z4, z8, 0);
#else
  __builtin_amdgcn_tensor_load_to_lds(g0, g1, z4, z4, 0);
#endif
}

// ---------------------------------------------------------------------------
// GEMM1: xz[m, n] = sum_k (x[b,k,t] * mask[b,k,t]) * W_in[k, n]
//   m = b*SEQ + t, M = 65536, N = 1024, K = 64 (16 WMMA k-steps)
//   A loads coalesced along t; x/mask (33 MB) are L2-resident across n-tiles.
// ---------------------------------------------------------------------------
__global__ void __launch_bounds__(128)
k_gemm_xz(const float* __restrict__ x, const float* __restrict__ mask,
          const float* __restrict__ Win, float* __restrict__ xz) {
  const int lane  = threadIdx.x & 31;
  const int wave  = threadIdx.x >> 5;
  const int mTile = blockIdx.y;                 // 0..4095
  const int nTile = blockIdx.x * 4 + wave;      // 0..63
  const int mRow  = mTile * 16 + (lane & 15);
  const int b     = mRow >> 11;
  const int t     = mRow & (SEQ - 1);
  const int n     = nTile * 16 + (lane & 15);
  const int khalf = (lane >> 4) * 2;

  v8f c = {};
#pragma unroll 4
  for (int ks = 0; ks < 16; ++ks) {
    v2f a, bm;
#pragma unroll
    for (int j = 0; j < 2; ++j) {
      const int k  = ks * 4 + khalf + j;
      const int xi = (b * DMODEL + k) * SEQ + t;   // lanes 0-15: contiguous t
      a[j]  = x[xi] * mask[xi];
      bm[j] = Win[k * NXZ + n];                    // lanes 0-15: contiguous n
    }
    c = wmma4(a, bm, c);
  }
  const int rbase = mTile * 16 + 8 * (lane >> 4);
#pragma unroll
  for (int j = 0; j < 8; ++j)
    xz[(size_t)(rbase + j) * NXZ + n] = c[j];
}

// ---------------------------------------------------------------------------
// Causal depthwise conv(4) + bias + SiLU over the xin half of xz
// ---------------------------------------------------------------------------
__global__ void __launch_bounds__(256)
k_conv_silu(const float* __restrict__ xz, const float* __restrict__ cw,
            const float* __restrict__ cb, float* __restrict__ xin) {
  const int idx = blockIdx.x * 256 + threadIdx.x;  // over MTOT*DINNER
  const int d   = idx & (DINNER - 1);
  const int m   = idx >> 9;
  const int t   = m & (SEQ - 1);
  float acc = cb[d];
#pragma unroll
  for (int j = 0; j < DCONV; ++j) {
    const int ti = t + j - (DCONV - 1);
    if (ti >= 0)
      acc += cw[d * DCONV + j] * xz[(size_t)(m + ti - t) * NXZ + d];
  }
  xin[(size_t)m * DINNER + d] = acc / (1.0f + __expf(-acc));  // silu
}

// ---------------------------------------------------------------------------
// GEMM2: dbc[m, n] = sum_k xin[m,k] * W_x[k, n]   (N padded 44 -> 48)
//   M = 65536, K = 512 (128 WMMA k-steps), 3 n-tiles per block (3 waves).
//   A-tile (16x512 f32) DMA'd once per block into LDS by the TDM (wave 0),
//   with padding so strided per-lane reads are bank-conflict-free.
// ---------------------------------------------------------------------------
__global__ void __launch_bounds__(96)
k_gemm_dbc(const float* __restrict__ xin, const float* __restrict__ Wx,
           float* __restrict__ dbc) {
  __shared__ float sA[16 * 516];   // 516 = 512 + 2 pad dwords per 256 stored
  const int lane  = threadIdx.x & 31;
  const int nTile = threadIdx.x >> 5;  // 0..2
  const int mTile = blockIdx.x;

  if (threadIdx.x < 32) {          // TDM issued once (wave 0 only, execz skip)
    tdm_load_tile_16x512(xin + (size_t)mTile * 16 * DINNER, sA);
    __builtin_amdgcn_s_wait_tensorcnt(0);
  }
  __syncthreads();

  const int row   = lane & 15;
  const int n     = nTile * 16 + row;
  const int khalf = (lane >> 4) * 2;

  v8f c = {};
  for (int ks = 0; ks < DINNER / 4; ++ks) {
    v2f a, bm;
#pragma unroll
    for (int j = 0; j < 2; ++j) {
      const int k = ks * 4 + khalf + j;
      a[j]  = sA[row * 516 + k + ((k >> 8) << 1)];
      bm[j] = (n < 44) ? Wx[k * 44 + n] : 0.0f;
    }
    c = wmma4(a, bm, c);
  }
  const int rbase = mTile * 16 + 8 * (lane >> 4);
#pragma unroll
  for (int j = 0; j < 8; ++j)
    dbc[(size_t)(rbase + j) * NDBC + n] = c[j];
}

// ---------------------------------------------------------------------------
// GEMM3 + epilogue: delta = softplus(dbc[:, :4] @ W_dt + b_dt)
//   K = 4 -> a single WMMA per tile. M = 65536, N = 512.
// ---------------------------------------------------------------------------
__global__ void __launch_bounds__(256)
k_gemm_delta(const float* __restrict__ dbc, const float* __restrict__ Wdt,
             const float* __restrict__ bdt, float* __restrict__ delta) {
  const int lane  = threadIdx.x & 31;
  const int wave  = threadIdx.x >> 5;           // 0..7
  const int mTile = blockIdx.y;
  const int nTile = blockIdx.x * 8 + wave;      // 0..31
  const int mRow  = mTile * 16 + (lane & 15);
  const int n     = nTile * 16 + (lane & 15);
  const int khalf = (lane >> 4) * 2;

  v2f a, bm;
#pragma unroll
  for (int j = 0; j < 2; ++j) {
    const int k = khalf + j;
    a[j]  = dbc[(size_t)mRow * NDBC + k];
    bm[j] = Wdt[k * DINNER + n];
  }
  v8f c = {};
  c = wmma4(a, bm, c);

  const int rbase = mTile * 16 + 8 * (lane >> 4);
  const float bias = bdt[n];
#pragma unroll
  for (int j = 0; j < 8; ++j) {
    const float v  = c[j] + bias;
    const float sp = (v > 20.0f) ? v : log1pf(__expf(v));  // softplus
    delta[(size_t)(rbase + j) * DINNER + n] = sp;
  }
}

// ---------------------------------------------------------------------------
// Selective scan fused with D-skip, SiLU gate and sequence mean:
//   acc[b,d] = (1/SEQ) * sum_t (y_t + D[d]*x_t) * silu(z_t)
// One thread per (b,d) channel, 20-wide state in registers.  Per chunk of 16
// timesteps: B/C vectors staged in LDS (shared by the 256 d of a block) and
// the thread's delta/x/z batch-loaded into registers so all 48 loads issue in
// one clause, hiding latency across the sequential exp->fma chain.
// ---------------------------------------------------------------------------
__global__ void __launch_bounds__(256)
k_scan(const float* __restrict__ xin, const float* __restrict__ delta,
       const float* __restrict__ dbc, const float* __restrict__ xz,
       const float* __restrict__ Alog, const float* __restrict__ Dp,
       float* __restrict__ acc) {
  const int b = blockIdx.x >> 1;
  const int d = ((blockIdx.x & 1) << 8) + threadIdx.x;

  __shared__ float sBC[16 * 40];  // 16 timesteps x (B[20] | C[20])

  float A[DSTATE], h[DSTATE];
#pragma unroll
  for (int nn = 0; nn < DSTATE; ++nn) {
    A[nn] = -__expf(Alog[d * DSTATE + nn]);
    h[nn] = 0.0f;
  }
  const float Dd = Dp[d];
  float s = 0.0f;

  for (int t0 = 0; t0 < SEQ; t0 += 16) {
    const int base = b * SEQ + t0;
    for (int i = threadIdx.x; i < 16 * 40; i += 256) {
      const int tt = i / 40, j = i - tt * 40;
      sBC[i] = dbc[(size_t)(base + tt) * NDBC + DTRANK + j];
    }
    // hint next chunk's streams toward the caches (speculative)
    if (t0 + 16 < SEQ) {
      const size_t mn = (size_t)(base + 16);
      __builtin_prefetch(&delta[mn * DINNER + d], 0, 1);
      __builtin_prefetch(&xin[mn * DINNER + d], 0, 1);
      __builtin_prefetch(&xz[mn * NXZ + DINNER + d], 0, 1);
    }
    float dl[16], xv[16], zv[16];
#pragma unroll
    for (int tt = 0; tt < 16; ++tt) {
      const size_t m = (size_t)(base + tt);
      dl[tt] = delta[m * DINNER + d];
      xv[tt] = xin[m * DINNER + d];
      zv[tt] = xz[m * NXZ + DINNER + d];
    }
    __syncthreads();
#pragma unroll 2
    for (int tt = 0; tt < 16; ++tt) {
      const float dlt = dl[tt];
      const float dx  = dlt * xv[tt];
      const float* Bt = &sBC[tt * 40];
      float y = 0.0f;
#pragma unroll
      for (int nn = 0; nn < DSTATE; ++nn) {
        const float dA = __expf(dlt * A[nn]);
        h[nn] = dA * h[nn] + dx * Bt[nn];
        y += h[nn] * Bt[20 + nn];
      }
      const float z = zv[tt];
      s += (y + Dd * xv[tt]) * (z / (1.0f + __expf(-z)));
    }
    __syncthreads();
  }
  acc[b * DINNER + d] = s * (1.0f / SEQ);
}

// ---------------------------------------------------------------------------
// Head: ts = acc @ W_out ; static MLP ; concat ; 3 small layers -> (32, 2)
// (mean over SEQ commutes with W_out, so ts = (mean_t ycomb) @ W_out)
// ---------------------------------------------------------------------------
__global__ void __launch_bounds__(256)
k_head(const float* __restrict__ acc, const float* __restrict__ Wout,
       const float* __restrict__ stat,
       const float* __restrict__ Ws1, const float* __restrict__ bs1,
       const float* __restrict__ Ws2, const float* __restrict__ bs2,
       const float* __restrict__ Wc1, const float* __restrict__ bc1,
       const float* __restrict__ Wc2, const float* __restrict__ bc2,
       const float* __restrict__ Wc3, const float* __restrict__ bc3,
       float* __restrict__ out) {
  const int bb  = blockIdx.x;
  const int tid = threadIdx.x;
  __shared__ float sC[192];   // [ts(64) | h2(128)]
  __shared__ float sH1[256];
  __shared__ float sC1[32];
  __shared__ float sC2[16];

  if (tid < 64) {  // ts row
    float v = 0.0f;
    for (int k = 0; k < DINNER; ++k)
      v += acc[bb * DINNER + k] * Wout[k * DMODEL + tid];
    sC[tid] = v;
  }
  {  // h1 = relu(static @ W_s1 + b_s1)
    float v = bs1[tid];
    for (int k = 0; k < 128; ++k) v += stat[bb * 128 + k] * Ws1[k * 256 + tid];
    sH1[tid] = fmaxf(v, 0.0f);
  }
  __syncthreads();
  if (tid < 128) {  // h2 = relu(h1 @ W_s2 + b_s2)
    float v = bs2[tid];
    for (int k = 0; k < 256; ++k) v += sH1[k] * Ws2[k * 128 + tid];
    sC[64 + tid] = fmaxf(v, 0.0f);
  }
  __syncthreads();
  if (tid < 32) {
    float v = bc1[tid];
    for (int k = 0; k < 192; ++k) v += sC[k] * Wc1[k * 32 + tid];
    sC1[tid] = fmaxf(v, 0.0f);
  }
  __syncthreads();
  if (tid < 16) {
    float v = bc2[tid];
    for (int k = 0; k < 32; ++k) v += sC1[k] * Wc2[k * 16 + tid];
    sC2[tid] = fmaxf(v, 0.0f);
  }
  __syncthreads();
  if (tid < 2) {
    float v = bc3[tid];
    for (int k = 0; k < 16; ++k) v += sC2[k] * Wc3[k * 2 + tid];
    out[bb * 2 + tid] = v;
  }
}

// ---------------------------------------------------------------------------
extern "C" void kernel_launch(void* const* d_in, const int* in_sizes, int n_in,
                              void* d_out, int out_size, void* d_ws,
                              size_t ws_size, hipStream_t stream) {
  (void)in_sizes; (void)n_in; (void)out_size; (void)ws_size;
  // dict order: x, static, time, sensor_mask, then params in insertion order
  const float* x    = (const float*)d_in[0];
  const float* stat = (const float*)d_in[1];
  /* d_in[2] = time (unused by reference) */
  const float* mask = (const float*)d_in[3];
  const float* Win  = (const float*)d_in[4];
  const float* cw   = (const float*)d_in[5];
  const float* cb   = (const float*)d_in[6];
  const float* Wx   = (const float*)d_in[7];
  const float* Wdt  = (const float*)d_in[8];
  const float* bdt  = (const float*)d_in[9];
  const float* Alog = (const float*)d_in[10];
  const float* Dp   = (const float*)d_in[11];
  const float* Wout = (const float*)d_in[12];
  const float* Ws1  = (const float*)d_in[13];
  const float* bs1  = (const float*)d_in[14];
  const float* Ws2  = (const float*)d_in[15];
  const float* bs2  = (const float*)d_in[16];
  const float* Wc1  = (const float*)d_in[17];
  const float* bc1  = (const float*)d_in[18];
  const float* Wc2  = (const float*)d_in[19];
  const float* bc2  = (const float*)d_in[20];
  const float* Wc3  = (const float*)d_in[21];
  const float* bc3  = (const float*)d_in[22];

  char* ws = (char*)d_ws;
  float* xz    = (float*)ws; ws += (size_t)MTOT * NXZ    * sizeof(float); // 256 MB
  float* xin   = (float*)ws; ws += (size_t)MTOT * DINNER * sizeof(float); // 128 MB
  float* dbc   = (float*)ws; ws += (size_t)MTOT * NDBC   * sizeof(float); //  12 MB
  float* delta = (float*)ws; ws += (size_t)MTOT * DINNER * sizeof(float); // 128 MB
  float* acc   = (float*)ws; ws += (size_t)BSZ  * DINNER * sizeof(float);

  k_gemm_xz   <<<dim3(16, MTOT / 16), 128, 0, stream>>>(x, mask, Win, xz);
  k_conv_silu <<<(MTOT * DINNER) / 256, 256, 0, stream>>>(xz, cw, cb, xin);
  k_gemm_dbc  <<<MTOT / 16, 96, 0, stream>>>(xin, Wx, dbc);
  k_gemm_delta<<<dim3(4, MTOT / 16), 256, 0, stream>>>(dbc, Wdt, bdt, delta);
  k_scan      <<<BSZ * 2, 256, 0, stream>>>(xin, delta, dbc, xz, Alog, Dp, acc);
  k_head      <<<BSZ, 256, 0, stream>>>(acc, Wout, stat, Ws1, bs1, Ws2, bs2,
                                        Wc1, bc1, Wc2, bc2, Wc3, bc3,
                                        (float*)d_out);
}